// ImprovedMultiHeadAttention_17669495456304
// MI455X (gfx1250) — compile-verified
//
#include <hip/hip_runtime.h>

// Problem constants (from reference): E = H = N = 64, L = 256, head_dim = 1.
#define LL 256
#define EE 64
#define LN_EPS 1e-5f

typedef float v2f __attribute__((ext_vector_type(2)));
typedef float v4f __attribute__((ext_vector_type(4)));
typedef float v8f __attribute__((ext_vector_type(8)));

__device__ __forceinline__ v8f wmma_f32(v2f a, v2f b, v8f c) {
    // D = A(16x4) * B(4x16) + C(16x16), all f32, wave32
    return __builtin_amdgcn_wmma_f32_16x16x4_f32(
        /*neg_a=*/false, a, /*neg_b=*/false, b,
        /*c_mod=*/(short)0, c, /*reuse_a=*/false, /*reuse_b=*/false);
}

// ---------------------------------------------------------------------------
// Kernel 1: one block per batch n. Computes z[n,l,e] = LayerNorm_e(y[n,l,e])
// where y = (softmax(Q_n^T K_n) @ V_n)^T @ W_out^T + b_out.
// The residual +query[n,l,h] cancels in the LN centering (constant over e),
// so z is independent of h.
// ---------------------------------------------------------------------------
__global__ __launch_bounds__(256)
void attn_fused_kernel(const float* __restrict__ values,  // [64,256,64]
                       const float* __restrict__ keys,    // [64,256,64]
                       const float* __restrict__ query,   // [64,256,64]
                       const float* __restrict__ W_out,   // [64,64]
                       const float* __restrict__ b_out,   // [64]
                       const float* __restrict__ ln_w,    // [64]
                       const float* __restrict__ ln_b,    // [64]
                       float* __restrict__ z)             // [64,256,64] ws
{
    __shared__ float lds0[EE * LL];  // out2[e*256 + l]          (64 KB)
    __shared__ float lds1[EE * LL];  // energy/attn, then y[l*64+e] (64 KB)

    const int n    = blockIdx.x;
    const int tid  = threadIdx.x;
    const int w    = tid >> 5;       // wave id 0..7
    const int lane = tid & 31;
    const int half = lane >> 4;      // 0: lanes 0-15, 1: lanes 16-31
    const int lm   = lane & 15;
    const int kb   = half * 2;       // K-base within 4-wide K slab (A/B layout)

    const float* Q = query + n * (LL * EE);
    const float* K = keys  + n * (LL * EE);

    // ---- Stage 1: energy[a,b] = sum_l Q[l,a] * K[l,b]   (64x64, Kdim=256)
    // 16 tiles of 16x16; 2 tiles per wave.
    for (int t = w * 2; t < w * 2 + 2; ++t) {
        const int i = t >> 2;        // a-tile
        const int j = t & 3;         // b-tile
        const int acol = i * 16 + lm;
        const int bcol = j * 16 + lm;
        v8f c = {};
        for (int l0 = 0; l0 < LL; l0 += 4) {
            v2f a, b;
            a.x = Q[(l0 + kb    ) * EE + acol];
            a.y = Q[(l0 + kb + 1) * EE + acol];
            b.x = K[(l0 + kb    ) * EE + bcol];
            b.y = K[(l0 + kb + 1) * EE + bcol];
            c = wmma_f32(a, b, c);
        }
        #pragma unroll
        for (int r = 0; r < 8; ++r)
            lds1[(i * 16 + half * 8 + r) * EE + j * 16 + lm] = c[r];
    }
    __syncthreads();

    // ---- Stage 2: softmax over b for each row a (scale = 1/sqrt(1) = 1)
    if (tid < 64) {
        float* row = lds1 + tid * EE;
        float m = row[0];
        for (int b = 1; b < EE; ++b) m = fmaxf(m, row[b]);
        float s = 0.f;
        for (int b = 0; b < EE; ++b) { float e = expf(row[b] - m); row[b] = e; s += e; }
        const float inv = 1.f / s;
        for (int b = 0; b < EE; ++b) row[b] *= inv;
    }
    __syncthreads();

    // ---- Stage 3: out2[e,l] = sum_k attn[e,k] * values[k,l,n]  (64x256, K=64)
    // 64 tiles; 8 per wave.
    for (int t = w * 8; t < w * 8 + 8; ++t) {
        const int i = t >> 4;        // e-tile 0..3
        const int j = t & 15;        // l-tile 0..15
        const int arow = (i * 16 + lm) * EE;
        const int bcol = (j * 16 + lm) * EE + n;   // values[., l, n]
        v8f c = {};
        for (int k0 = 0; k0 < EE; k0 += 4) {
            v2f a, b;
            a.x = lds1[arow + k0 + kb];
            a.y = lds1[arow + k0 + kb + 1];
            b.x = values[(k0 + kb    ) * (LL * EE) + bcol];
            b.y = values[(k0 + kb + 1) * (LL * EE) + bcol];
            c = wmma_f32(a, b, c);
        }
        #pragma unroll
        for (int r = 0; r < 8; ++r)
            lds0[(i * 16 + half * 8 + r) * LL + j * 16 + lm] = c[r];
    }
    __syncthreads();

    // ---- Stage 4: y[l,e] = sum_{e'} out2[e',l] * W_out[e,e'] + b_out[e]
    // (256x64, K=64). 64 tiles; 8 per wave. Overwrites lds1 (attn dead now).
    for (int t = w * 8; t < w * 8 + 8; ++t) {
        const int i = t >> 2;        // l-tile 0..15
        const int j = t & 3;         // e-tile 0..3
        const int wrow = (j * 16 + lm) * EE;       // W_out[e, .]
        v8f c = {};
        for (int k0 = 0; k0 < EE; k0 += 4) {
            v2f a, b;
            a.x = lds0[(k0 + kb    ) * LL + i * 16 + lm];
            a.y = lds0[(k0 + kb + 1) * LL + i * 16 + lm];
            b.x = W_out[wrow + k0 + kb];
            b.y = W_out[wrow + k0 + kb + 1];
            c = wmma_f32(a, b, c);
        }
        const float bias = b_out[j * 16 + lm];
        #pragma unroll
        for (int r = 0; r < 8; ++r)
            lds1[(i * 16 + half * 8 + r) * EE + j * 16 + lm] = c[r] + bias;
    }
    __syncthreads();

    // ---- Stage 5: LayerNorm over e for each row l (in place in lds1)
    {
        float* row = lds1 + tid * EE;     // tid = l, 256 rows
        float s = 0.f, sq = 0.f;
        for (int e = 0; e < EE; ++e) { float v = row[e]; s += v; sq += v * v; }
        const float mu   = s * (1.f / EE);
        const float var  = sq * (1.f / EE) - mu * mu;
        const float rstd = rsqrtf(var + LN_EPS);
        for (int e = 0; e < EE; ++e)
            row[e] = (row[e] - mu) * rstd * ln_w[e] + ln_b[e];
    }
    __syncthreads();

    // ---- Stage 6: write z[n] (16384 floats) to workspace, vectorized
    {
        v4f* __restrict__ z4 = (v4f*)(z + n * (LL * EE));
        const v4f* l4 = (const v4f*)lds1;
        for (int idx = tid; idx < (LL * EE) / 4; idx += 256)
            z4[idx] = l4[idx];
    }
}

// ---------------------------------------------------------------------------
// Kernel 2: broadcast z (4 MB, L2 resident) into the 64 h-copies of the
// 256 MB output with non-temporal b128 streaming stores.
// out[h,n,l,e] = z[n,l,e];  h-stride = 262144 float4.
// ---------------------------------------------------------------------------
__global__ __launch_bounds__(256)
void broadcast_store_kernel(const float* __restrict__ zf, float* __restrict__ outf)
{
    const v4f* __restrict__ z4 = (const v4f*)zf;
    v4f* __restrict__ out4 = (v4f*)outf;

    const unsigned gid = blockIdx.x * 256u + threadIdx.x;  // 0 .. 4,194,303
    const unsigned s   = gid & 262143u;                    // src float4 index
    const unsigned h0  = gid >> 18;                        // 0..15
    const v4f v = z4[s];
    #pragma unroll
    for (int k = 0; k < 4; ++k) {
        const size_t h = (size_t)(h0 + 16u * k);
        __builtin_nontemporal_store(v, out4 + h * 262144u + s);
    }
}

extern "C" void kernel_launch(void* const* d_in, const int* in_sizes, int n_in,
                              void* d_out, int out_size, void* d_ws, size_t ws_size,
                              hipStream_t stream) {
    const float* values = (const float*)d_in[0];
    const float* keys   = (const float*)d_in[1];
    const float* query  = (const float*)d_in[2];
    const float* W_out  = (const float*)d_in[3];
    const float* b_out  = (const float*)d_in[4];
    const float* ln_w   = (const float*)d_in[5];
    const float* ln_b   = (const float*)d_in[6];
    float* out = (float*)d_out;                 // [64,64,256,64]
    float* z   = (float*)d_ws;                  // 4 MB scratch: z[n,l,e]

    attn_fused_kernel<<<64, 256, 0, stream>>>(values, keys, query,
                                              W_out, b_out, ln_w, ln_b, z);
    broadcast_store_kernel<<<16384, 256, 0, stream>>>(z, out);
}